// MixHop_77129022701598
// MI455X (gfx1250) — compile-verified
//
#include <hip/hip_runtime.h>
#include <hip/hip_bf16.h>

// MixHop GNN forward for MI455X (gfx1250).
// Dense GEMMs on V_WMMA_F32_16X16X4_F32 (fp32 matrix path, wave32);
// sparse A*x as self-loop-init + wave-per-edge fp32 atomic scatter (L2-resident).
// Workspace requirement: (N + E + 4*N*180) floats ~= 146 MB.

#define NNODES 50000
#define NEDGES 400000
#define DIN    128
#define HID    60
#define NCLS   40
#define F3H    180
#define BNEPS  1e-5f

typedef float v2f __attribute__((ext_vector_type(2)));
typedef float v8f __attribute__((ext_vector_type(8)));

// ---------------- gcn_norm ----------------

__global__ void k_deg_init(float* __restrict__ deg) {
  int i = blockIdx.x * blockDim.x + threadIdx.x;
  if (i < NNODES) deg[i] = 1.0f;  // self-loop contributes 1 to every node's degree
}

__global__ void k_deg_accum(const int* __restrict__ col, float* __restrict__ deg) {
  int e = blockIdx.x * blockDim.x + threadIdx.x;
  if (e < NEDGES)
    __hip_atomic_fetch_add(&deg[col[e]], 1.0f, __ATOMIC_RELAXED, __HIP_MEMORY_SCOPE_AGENT);
}

__global__ void k_dinv(float* __restrict__ deg) {
  int i = blockIdx.x * blockDim.x + threadIdx.x;
  if (i < NNODES) deg[i] = rsqrtf(deg[i]);  // deg >= 1 always (self loops)
}

__global__ void k_norm(const int* __restrict__ row, const int* __restrict__ col,
                       const float* __restrict__ dis, float* __restrict__ nrm) {
  int e = blockIdx.x * blockDim.x + threadIdx.x;
  if (e < NEDGES) nrm[e] = dis[row[e]] * dis[col[e]];
}

// ---------------- sparse propagation: out = A_hat * hin ----------------

// out[i][:] = dis[i]^2 * hin[i][:]   (self-loop term; also zero-initializes accumulator)
__global__ void k_selfloop_init(const float* __restrict__ hin, const float* __restrict__ dis,
                                float* __restrict__ out, int F) {
  int i = blockIdx.x;
  float d = dis[i];
  float d2 = d * d;
  int f = threadIdx.x;
  if (f < F) out[(size_t)i * F + f] = d2 * hin[(size_t)i * F + f];
}

// out[col[e]][:] += nrm[e] * hin[row[e]][:]   (one wave32 per edge, coalesced feature strip)
__global__ void k_spmm_edges(const int* __restrict__ row, const int* __restrict__ col,
                             const float* __restrict__ nrm, const float* __restrict__ hin,
                             float* __restrict__ out, int F) {
  int e = blockIdx.x * (blockDim.x >> 5) + (threadIdx.x >> 5);
  if (e >= NEDGES) return;
  int lane = threadIdx.x & 31;
  int r = row[e], c = col[e];
  float w = nrm[e];
  const float* __restrict__ src = hin + (size_t)r * F;
  float* __restrict__ dst = out + (size_t)c * F;
  for (int f = lane; f < F; f += 32)
    __hip_atomic_fetch_add(dst + f, w * src[f], __ATOMIC_RELAXED, __HIP_MEMORY_SCOPE_AGENT);
}

// ---------------- dense: WMMA fp32 16x16x4 tile ----------------
// A fragment: lane L holds hin[mbase + (L&15)][k + 2*(L>>4) .. +1]  (8B contiguous load)
// B fragment: lane L holds W[nbase + (L&15)][k + 2*(L>>4) .. +1]    (out = h * W^T)
// C/D: VGPR v <-> row mbase + v + 8*(L>>4), col nbase + (L&15)
__device__ __forceinline__ v8f wmma_tile_f32(const float* __restrict__ arow,
                                             const float* __restrict__ brow, int F) {
  v8f acc = {};
#pragma unroll 4
  for (int k = 0; k < F; k += 4) {
    v2f a = *(const v2f*)(arow + k);
    v2f b = *(const v2f*)(brow + k);
    acc = __builtin_amdgcn_wmma_f32_16x16x4_f32(false, a, false, b, (short)0, acc,
                                                false, false);
  }
  return acc;
}

// One MixHop layer's three GEMMs + bias + eval-mode BatchNorm, fused.
// Tile id space: 3 powers x 4 col-tiles(64 cols, mask >=60) x 3125 row-tiles.
__global__ void k_mixhop_gemm(const float* __restrict__ h0, const float* __restrict__ h1,
                              const float* __restrict__ h2,
                              const float* __restrict__ W,    // [3, HID, F]
                              const float* __restrict__ bias, // [3, HID]
                              const float* __restrict__ bng, const float* __restrict__ bnb,
                              const float* __restrict__ bnm, const float* __restrict__ bnv,
                              float* __restrict__ out,        // [N, 180]
                              int F) {
  const int NT_ROW = NNODES / 16;          // 3125 exact
  const int TILES = 3 * 4 * NT_ROW;
  int wave = blockIdx.x * (blockDim.x >> 5) + (threadIdx.x >> 5);
  if (wave >= TILES) return;
  int p = wave / (4 * NT_ROW);
  int r = wave % (4 * NT_ROW);
  int ctile = r & 3;
  int mbase = (r >> 2) * 16;

  const float* __restrict__ hin = (p == 0) ? h0 : (p == 1) ? h1 : h2;
  const float* __restrict__ Wp = W + (size_t)p * HID * F;

  int lane = threadIdx.x & 31;
  int l15 = lane & 15;
  int half = lane >> 4;
  int nl = ctile * 16 + l15;               // local col in [0,64)
  int nc = (nl < HID) ? nl : (HID - 1);    // clamp loads; masked cols never stored

  const float* arow = hin + (size_t)(mbase + l15) * F + 2 * half;
  const float* brow = Wp + (size_t)nc * F + 2 * half;
  v8f acc = wmma_tile_f32(arow, brow, F);

  if (nl < HID) {
    int g = p * HID + nl;                  // global output column [0,180)
    float sc = bng[g] * rsqrtf(bnv[g] + BNEPS);
    float sh = bnb[g] - bnm[g] * sc;
    float bs = bias[p * HID + nl];
    float* orow = out + (size_t)(mbase + 8 * half) * F3H + g;
#pragma unroll
    for (int v = 0; v < 8; ++v)
      orow[(size_t)v * F3H] = (acc[v] + bs) * sc + sh;
  }
}

// Final linear: [N,180] x lin_W[40,180]^T + b.  3 col-tiles (48 cols, mask >=40).
__global__ void k_final_gemm(const float* __restrict__ hin,
                             const float* __restrict__ Wl,   // [NCLS, 180]
                             const float* __restrict__ bl,   // [NCLS]
                             float* __restrict__ out) {      // [N, NCLS]
  const int NT_ROW = NNODES / 16;
  const int TILES = 3 * NT_ROW;
  int wave = blockIdx.x * (blockDim.x >> 5) + (threadIdx.x >> 5);
  if (wave >= TILES) return;
  int ctile = wave % 3;
  int mbase = (wave / 3) * 16;

  int lane = threadIdx.x & 31;
  int l15 = lane & 15;
  int half = lane >> 4;
  int nl = ctile * 16 + l15;               // [0,48)
  int nc = (nl < NCLS) ? nl : (NCLS - 1);

  const float* arow = hin + (size_t)(mbase + l15) * F3H + 2 * half;
  const float* brow = Wl + (size_t)nc * F3H + 2 * half;
  v8f acc = wmma_tile_f32(arow, brow, F3H);

  if (nl < NCLS) {
    float bs = bl[nl];
    float* orow = out + (size_t)(mbase + 8 * half) * NCLS + nl;
#pragma unroll
    for (int v = 0; v < 8; ++v)
      orow[(size_t)v * NCLS] = acc[v] + bs;
  }
}

// ---------------- launch ----------------

extern "C" void kernel_launch(void* const* d_in, const int* in_sizes, int n_in,
                              void* d_out, int out_size, void* d_ws, size_t ws_size,
                              hipStream_t stream) {
  const float* x   = (const float*)d_in[0];
  const int*   ei  = (const int*)d_in[1];
  const int*   row = ei;            // edge_index[0]
  const int*   col = ei + NEDGES;   // edge_index[1]
  const float* c1W = (const float*)d_in[2];
  const float* c1b = (const float*)d_in[3];
  const float* c2W = (const float*)d_in[4];
  const float* c2b = (const float*)d_in[5];
  const float* c3W = (const float*)d_in[6];
  const float* c3b = (const float*)d_in[7];
  const float* bn1g = (const float*)d_in[8],  *bn1b = (const float*)d_in[9];
  const float* bn1m = (const float*)d_in[10], *bn1v = (const float*)d_in[11];
  const float* bn2g = (const float*)d_in[12], *bn2b = (const float*)d_in[13];
  const float* bn2m = (const float*)d_in[14], *bn2v = (const float*)d_in[15];
  const float* bn3g = (const float*)d_in[16], *bn3b = (const float*)d_in[17];
  const float* bn3m = (const float*)d_in[18], *bn3v = (const float*)d_in[19];
  const float* linW = (const float*)d_in[20];
  const float* linb = (const float*)d_in[21];
  float* outp = (float*)d_out;

  float* ws = (float*)d_ws;
  float* deg = ws;                         // [N] -> becomes dis (rsqrt of degree)
  float* nrm = deg + NNODES;               // [E]
  float* Q = nrm + NEDGES;                 // [N,180]
  float* R = Q + (size_t)NNODES * F3H;     // [N,180]
  float* P = R + (size_t)NNODES * F3H;     // [N,180]
  float* S = P + (size_t)NNODES * F3H;     // [N,180]

  const int NB_N = (NNODES + 255) / 256;
  const int NB_E = (NEDGES + 255) / 256;
  const int NB_EW = (NEDGES + 7) / 8;                       // 8 waves/block, wave per edge
  const int NB_MH = (3 * 4 * (NNODES / 16) + 7) / 8;        // mixhop gemm tiles
  const int NB_FG = (3 * (NNODES / 16) + 7) / 8;            // final gemm tiles

  // gcn_norm
  k_deg_init<<<NB_N, 256, 0, stream>>>(deg);
  k_deg_accum<<<NB_E, 256, 0, stream>>>(col, deg);
  k_dinv<<<NB_N, 256, 0, stream>>>(deg);
  k_norm<<<NB_E, 256, 0, stream>>>(row, col, deg, nrm);

  // layer 1 (F = 128): Q = A*x, R = A*Q, S = BN1(concat GEMMs)
  k_selfloop_init<<<NNODES, 128, 0, stream>>>(x, deg, Q, DIN);
  k_spmm_edges<<<NB_EW, 256, 0, stream>>>(row, col, nrm, x, Q, DIN);
  k_selfloop_init<<<NNODES, 128, 0, stream>>>(Q, deg, R, DIN);
  k_spmm_edges<<<NB_EW, 256, 0, stream>>>(row, col, nrm, Q, R, DIN);
  k_mixhop_gemm<<<NB_MH, 256, 0, stream>>>(x, Q, R, c1W, c1b,
                                           bn1g, bn1b, bn1m, bn1v, S, DIN);

  // layer 2 (F = 180): in S -> out P
  k_selfloop_init<<<NNODES, 192, 0, stream>>>(S, deg, Q, F3H);
  k_spmm_edges<<<NB_EW, 256, 0, stream>>>(row, col, nrm, S, Q, F3H);
  k_selfloop_init<<<NNODES, 192, 0, stream>>>(Q, deg, R, F3H);
  k_spmm_edges<<<NB_EW, 256, 0, stream>>>(row, col, nrm, Q, R, F3H);
  k_mixhop_gemm<<<NB_MH, 256, 0, stream>>>(S, Q, R, c2W, c2b,
                                           bn2g, bn2b, bn2m, bn2v, P, F3H);

  // layer 3 (F = 180): in P -> out S
  k_selfloop_init<<<NNODES, 192, 0, stream>>>(P, deg, Q, F3H);
  k_spmm_edges<<<NB_EW, 256, 0, stream>>>(row, col, nrm, P, Q, F3H);
  k_selfloop_init<<<NNODES, 192, 0, stream>>>(Q, deg, R, F3H);
  k_spmm_edges<<<NB_EW, 256, 0, stream>>>(row, col, nrm, Q, R, F3H);
  k_mixhop_gemm<<<NB_MH, 256, 0, stream>>>(P, Q, R, c3W, c3b,
                                           bn3g, bn3b, bn3m, bn3v, S, F3H);

  // final linear: S -> d_out [N,40]
  k_final_gemm<<<NB_FG, 256, 0, stream>>>(S, linW, linb, outp);
}